// SymmetricKMeans_16226386444303
// MI455X (gfx1250) — compile-verified
//
#include <hip/hip_runtime.h>

// ---- problem constants (match reference) ----
#define R_      10
#define B_      8
#define NPG_    1024
#define M_      512
#define G_      (R_ * B_)          // 80 groups
#define TOL_    1e-3f
#define MAXIT_  300
#define NTHR_   512                // 16 wave32 waves per block
#define NWAVE_  (NTHR_ / 32)

typedef __attribute__((ext_vector_type(2))) float v2f;
typedef __attribute__((ext_vector_type(8))) float v8f;

// Branchless (cmp + cndmask) argmax merge with first-index tie-break.
__device__ __forceinline__ void argmax_merge(float& bv, int& bi, float ov, int oi) {
    const bool t = (ov > bv) | ((ov == bv) & (oi < bi));
    bv = t ? ov : bv;
    bi = t ? oi : bi;
}

__global__ __launch_bounds__(NTHR_)
void kmeans_persistent_kernel(const float* __restrict__ pos,
                              float* __restrict__ out)
{
    // LDS state for one group (~45 KB; WGP has 320 KB)
    __shared__ __align__(16) float s_pts[NPG_ * 4];   // points, xyz0 padded
    __shared__ __align__(16) float s_cent[M_ * 4];    // centroids, xyz0 padded
    __shared__ __align__(16) float s_c2neg[M_];       // -0.5 * |c|^2
    __shared__ float s_sum[M_ * 4];                   // scatter sums
    __shared__ int   s_cnt[M_];                       // scatter counts
    __shared__ int   s_cls[NPG_];                     // labels
    __shared__ float s_mind[NPG_];                    // FPS min-dist
    __shared__ float s_rv[NWAVE_];                    // per-wave reduction values
    __shared__ int   s_ri[NWAVE_];                    // per-wave reduction indices
    __shared__ int   s_flag;
    __shared__ int   s_last;

    const int g    = blockIdx.x;           // group id in [0, 80)
    const int tid  = threadIdx.x;
    const int wave = tid >> 5;             // 0..15
    const int lane = tid & 31;
    const int lhalf = lane >> 4;           // 0 or 1
    const int lmod  = lane & 15;
    const int ex  = g % B_;                // replicated example id
    const float* p = pos + (size_t)ex * NPG_ * 3;

    // ---- load points into LDS, padded to 4 components ----
    __builtin_prefetch(p + tid * 3, 0, 0);             // global_prefetch_b8
    for (int n = tid; n < NPG_; n += NTHR_) {
        s_pts[n * 4 + 0] = p[n * 3 + 0];
        s_pts[n * 4 + 1] = p[n * 3 + 1];
        s_pts[n * 4 + 2] = p[n * 3 + 2];
        s_pts[n * 4 + 3] = 0.0f;
        s_mind[n] = 3.4e38f;
    }
    if (tid == 0) {
        // deterministic pseudo-random FPS start per replica-group (LCG hash)
        s_last = (int)((1103515245u * (unsigned)(g + 1) + 12345u) % (unsigned)NPG_);
    }
    __syncthreads();

    // ---- farthest point sampling: pick M_ centroids ----
    for (int it = 0; it < M_; ++it) {
        const int last = s_last;
        if (tid < 4) s_cent[it * 4 + tid] = s_pts[last * 4 + tid];
        const float lx = s_pts[last * 4 + 0];
        const float ly = s_pts[last * 4 + 1];
        const float lz = s_pts[last * 4 + 2];
        float bv = -1.0f; int bi = 0;
#pragma unroll
        for (int k = 0; k < NPG_ / NTHR_; ++k) {       // 2 points / thread
            const int n = tid + k * NTHR_;
            const float dx = s_pts[n * 4 + 0] - lx;
            const float dy = s_pts[n * 4 + 1] - ly;
            const float dz = s_pts[n * 4 + 2] - lz;
            const float d  = dx * dx + dy * dy + dz * dz;
            const float mn = fminf(s_mind[n], d);
            s_mind[n] = mn;
            const bool t = mn > bv;
            bv = t ? mn : bv;
            bi = t ? n  : bi;
        }
        // wave-level argmax: 5 shuffle steps, no LDS, no barriers
#pragma unroll
        for (int d = 16; d > 0; d >>= 1)
            argmax_merge(bv, bi, __shfl_xor(bv, d), __shfl_xor(bi, d));
        if (lane == 0) { s_rv[wave] = bv; s_ri[wave] = bi; }
        __syncthreads();
        if (tid == 0) {
            float fv = s_rv[0]; int fi = s_ri[0];
            for (int w = 1; w < NWAVE_; ++w) argmax_merge(fv, fi, s_rv[w], s_ri[w]);
            s_last = fi;
        }
        __syncthreads();
    }

    // ---- k-means loop ----
    for (int iter = 0; iter < MAXIT_; ++iter) {
        // precompute -0.5*|c|^2, zero scatter buffers
        for (int m = tid; m < M_; m += NTHR_) {
            const float x = s_cent[m * 4 + 0];
            const float y = s_cent[m * 4 + 1];
            const float z = s_cent[m * 4 + 2];
            s_c2neg[m] = -0.5f * (x * x + y * y + z * z);
            s_sum[m * 4 + 0] = 0.0f;
            s_sum[m * 4 + 1] = 0.0f;
            s_sum[m * 4 + 2] = 0.0f;
            s_cnt[m] = 0;
        }
        if (tid == 0) s_flag = 0;
        __syncthreads();

        // ---- assign via WMMA f32 16x16x4:
        //   A = 16 centroids x K4, B = K4 x 16 points, C = -0.5|c|^2
        //   D[m][n] = c_m . p_n - 0.5|c_m|^2 ; argmax over m == nearest centroid.
        // Two point tiles per pass share the A operand and C seed (2 WMMA chains).
        // Lane L of D: point n = n0 + (L&15); centroids m = c0 + r + 8*(L>=16).
#pragma unroll
        for (int tt = 0; tt < 2; ++tt) {               // uniform over all waves
            const int t0  = wave + 32 * tt;
            const int np0 = t0 * 16 + lmod;
            const int np1 = np0 + 256;                 // tile t0+16
            const v2f b0 = *(const v2f*)&s_pts[np0 * 4 + 2 * lhalf];
            const v2f b1 = *(const v2f*)&s_pts[np1 * 4 + 2 * lhalf];
            float bv0 = -3.4e38f, bv1 = -3.4e38f;
            int   bi0 = 0,        bi1 = 0;
#pragma unroll 2
            for (int ct = 0; ct < M_ / 16; ++ct) {
                const int c0 = ct * 16;
                const v2f a = *(const v2f*)&s_cent[(c0 + lmod) * 4 + 2 * lhalf];
                v8f ci;
#pragma unroll
                for (int r = 0; r < 8; ++r)
                    ci[r] = s_c2neg[c0 + r + 8 * lhalf];
                v8f d0 = __builtin_amdgcn_wmma_f32_16x16x4_f32(
                        false, a, false, b0, (short)0, ci, false, false);
                v8f d1 = __builtin_amdgcn_wmma_f32_16x16x4_f32(
                        false, a, false, b1, (short)0, ci, false, false);
                // m ascends within a lane -> strict '>' keeps first index on ties
#pragma unroll
                for (int r = 0; r < 8; ++r) {
                    const int m = c0 + r + 8 * lhalf;
                    const bool t0b = d0[r] > bv0;
                    bv0 = t0b ? d0[r] : bv0;
                    bi0 = t0b ? m     : bi0;
                    const bool t1b = d1[r] > bv1;
                    bv1 = t1b ? d1[r] : bv1;
                    bi1 = t1b ? m     : bi1;
                }
            }
            // lanes L and L^16 hold the same point with disjoint centroid subsets
            argmax_merge(bv0, bi0, __shfl_xor(bv0, 16), __shfl_xor(bi0, 16));
            argmax_merge(bv1, bi1, __shfl_xor(bv1, 16), __shfl_xor(bi1, 16));
            if (lhalf == 0) {
                s_cls[np0] = bi0;
                s_cls[np1] = bi1;
            }
        }
        __syncthreads();

        // ---- scatter-mean update (LDS atomics) ----
        for (int n = tid; n < NPG_; n += NTHR_) {
            const int m = s_cls[n];
            atomicAdd(&s_sum[m * 4 + 0], s_pts[n * 4 + 0]);
            atomicAdd(&s_sum[m * 4 + 1], s_pts[n * 4 + 1]);
            atomicAdd(&s_sum[m * 4 + 2], s_pts[n * 4 + 2]);
            atomicAdd(&s_cnt[m], 1);
        }
        __syncthreads();

        for (int m = tid; m < M_; m += NTHR_) {
            const float cnt = (float)s_cnt[m];
            const bool  has = cnt > 0.0f;
            const float inv = has ? 1.0f / cnt : 0.0f;
            const float ox = s_cent[m * 4 + 0];
            const float oy = s_cent[m * 4 + 1];
            const float oz = s_cent[m * 4 + 2];
            const float nx = has ? s_sum[m * 4 + 0] * inv : ox;
            const float ny = has ? s_sum[m * 4 + 1] * inv : oy;
            const float nz = has ? s_sum[m * 4 + 2] * inv : oz;
            const float dx = nx - ox, dy = ny - oy, dz = nz - oz;
            if (dx * dx + dy * dy + dz * dz >= TOL_ * TOL_) s_flag = 1;  // benign race
            s_cent[m * 4 + 0] = nx;
            s_cent[m * 4 + 1] = ny;
            s_cent[m * 4 + 2] = nz;
        }
        __syncthreads();
        const int f = s_flag;
        __syncthreads();
        if (f == 0) break;      // uniform per block
    }

    // ---- score: L1 distance to assigned centroid, summed over group ----
    float acc = 0.0f;
    for (int n = tid; n < NPG_; n += NTHR_) {
        const int m = s_cls[n];
        acc += fabsf(s_pts[n * 4 + 0] - s_cent[m * 4 + 0])
             + fabsf(s_pts[n * 4 + 1] - s_cent[m * 4 + 1])
             + fabsf(s_pts[n * 4 + 2] - s_cent[m * 4 + 2]);
    }
#pragma unroll
    for (int d = 16; d > 0; d >>= 1)
        acc += __shfl_xor(acc, d);
    if (lane == 0) s_rv[wave] = acc;
    __syncthreads();

    // ---- write outputs: [classification | centroids | scores] flat ----
    float* out_cls = out;                              // G_*NPG_  = 81920
    float* out_c   = out + (size_t)G_ * NPG_;          // G_*M_*3  = 122880
    float* out_s   = out_c + (size_t)G_ * M_ * 3;      // G_       = 80
    for (int n = tid; n < NPG_; n += NTHR_)
        out_cls[(size_t)g * NPG_ + n] = (float)s_cls[n];
    for (int m = tid; m < M_; m += NTHR_) {
        out_c[((size_t)g * M_ + m) * 3 + 0] = s_cent[m * 4 + 0];
        out_c[((size_t)g * M_ + m) * 3 + 1] = s_cent[m * 4 + 1];
        out_c[((size_t)g * M_ + m) * 3 + 2] = s_cent[m * 4 + 2];
    }
    if (tid == 0) {
        float sc = 0.0f;
        for (int w = 0; w < NWAVE_; ++w) sc += s_rv[w];
        out_s[g] = sc;
    }
}

extern "C" void kernel_launch(void* const* d_in, const int* in_sizes, int n_in,
                              void* d_out, int out_size, void* d_ws, size_t ws_size,
                              hipStream_t stream)
{
    (void)in_sizes; (void)n_in; (void)out_size; (void)d_ws; (void)ws_size;
    const float* pos = (const float*)d_in[0];
    // d_in[1] (batch) is implied by the fixed group layout; unused.
    float* out = (float*)d_out;
    kmeans_persistent_kernel<<<G_, NTHR_, 0, stream>>>(pos, out);
}